// FVRF_41369124995663
// MI455X (gfx1250) — compile-verified
//
#include <hip/hip_runtime.h>

typedef __attribute__((ext_vector_type(2))) float v2f;
typedef __attribute__((ext_vector_type(8))) float v8f;

#define RR    64              // spatial rank
#define KM1   44              // harmonic coeffs minus baseline
#define NVOX  8192
#define LDC   (RR * KM1)      // 2816, leading dim of original C
#define NCOLS (KM1 * KM1)     // 1936 output cov columns
#define OUTROW 46             // K+1
#define OUTVOX (45 * 46)      // 2070 floats per voxel

constexpr int WAVES  = 2;              // waves per block
constexpr int BLOCK  = WAVES * 32;     // 64 threads
constexpr int MWAVE  = 64;             // output rows per wave (4 subtiles of 16)
constexpr int MBLOCK = WAVES * MWAVE;  // 128 rows per block
constexpr int KCHUNK = 64;             // == RR, so i = kc>>6 is constant per chunk
constexpr int NBLK   = 32;             // two 16-wide N tiles per wave/block
constexpr int NGRID  = (NCOLS + NBLK - 1) / NBLK;  // 61 (last block half-masked)

// ---------------------------------------------------------------------------
// Fused sandwich GEMM:  out_cov[v, a, b] = sum_{i,j} X[v,i] X[v,j] C[(i*44+a),(j*44+b)]
// as  P(8192x4096) @ Cp(4096x1936), P and Cp synthesized on the fly.
// Each wave: 64 rows x 32 cols (4 M-subtiles x 2 N-tiles, 8 f32 accumulators).
// ---------------------------------------------------------------------------
__global__ __launch_bounds__(BLOCK)
void cov_gemm(const float* __restrict__ X,
              const float* __restrict__ C,
              float* __restrict__ out)
{
    __shared__ float xs[MBLOCK][RR + 1];    // +1 pad: column reads conflict-free
    __shared__ float bs[KCHUNK][NBLK + 1];  // +1 pad: half-wave reads on disjoint banks

    const int tid  = threadIdx.x;
    const int wave = tid >> 5;
    const int lane = tid & 31;
    const int mblockbase = blockIdx.y * MBLOCK;
    const int n0         = blockIdx.x * NBLK;

    // Stage this block's X rows (128 x 64) into LDS.
    for (int idx = tid; idx < MBLOCK * RR; idx += BLOCK) {
        const int r = idx >> 6, c = idx & 63;
        xs[r][c] = X[(mblockbase + r) * RR + c];
    }

    // --- B-load assignment (fixed per thread): column nn = tid&31, rows j = (tid>>5)+2t
    const int nnld = tid & 31;
    int ncld = n0 + nnld; if (ncld > NCOLS - 1) ncld = NCOLS - 1;   // clamp tail block
    const int aald = ncld / KM1, bbld = ncld % KM1;
    const int jld0 = tid >> 5;

    // --- store-side column mapping (fixed per lane)
    const int c15   = lane & 15;
    const int ncol0 = n0 + c15;                 // always valid (n0 <= 1920)
    const int ncol1 = ncol0 + 16;               // may exceed 1935 in last block
    const int nc1c  = (ncol1 > NCOLS - 1) ? (NCOLS - 1) : ncol1;
    const long colofs0 = (long)(ncol0 / KM1 + 1) * OUTROW + (ncol0 % KM1) + 2;
    const long colofs1 = (long)(nc1c  / KM1 + 1) * OUTROW + (nc1c  % KM1) + 2;

    const int halfk = (lane >> 4) * 2;          // A/B frag: v0 holds K = 2*(lane>>4)
    const int rbase = wave * MWAVE + c15;

    v8f acc00 = {}, acc10 = {}, acc20 = {}, acc30 = {};
    v8f acc01 = {}, acc11 = {}, acc21 = {}, acc31 = {};

    for (int kc = 0; kc < RR * RR; kc += KCHUNK) {
        const int i = kc >> 6;                  // kron slot-1 index, constant per chunk

        __syncthreads();
        {   // Cooperative load of Cp chunk: bs[j][nn] = C[(i*44+a)*2816 + j*44 + b]
            const long rowb = (long)(i * KM1 + aald) * LDC + bbld;
            for (int j = jld0; j < KCHUNK; j += 2)
                bs[j][nnld] = C[rowb + (long)j * KM1];
        }
        __syncthreads();

        // Per-chunk constant A factors: X[row, i] for this lane's 4 row-subtiles.
        const float xi0 = xs[rbase +  0][i];
        const float xi1 = xs[rbase + 16][i];
        const float xi2 = xs[rbase + 32][i];
        const float xi3 = xs[rbase + 48][i];

        for (int k = 0; k < KCHUNK; k += 4) {
            const int j0 = k + halfk;

            v2f b0, b1;
            b0.x = bs[j0][c15];       b0.y = bs[j0 + 1][c15];
            b1.x = bs[j0][c15 + 16];  b1.y = bs[j0 + 1][c15 + 16];

            v2f a0, a1, a2, a3;
            a0.x = xi0 * xs[rbase +  0][j0];  a0.y = xi0 * xs[rbase +  0][j0 + 1];
            a1.x = xi1 * xs[rbase + 16][j0];  a1.y = xi1 * xs[rbase + 16][j0 + 1];
            a2.x = xi2 * xs[rbase + 32][j0];  a2.y = xi2 * xs[rbase + 32][j0 + 1];
            a3.x = xi3 * xs[rbase + 48][j0];  a3.y = xi3 * xs[rbase + 48][j0 + 1];

            acc00 = __builtin_amdgcn_wmma_f32_16x16x4_f32(false, a0, false, b0, (short)0, acc00, false, false);
            acc10 = __builtin_amdgcn_wmma_f32_16x16x4_f32(false, a1, false, b0, (short)0, acc10, false, false);
            acc20 = __builtin_amdgcn_wmma_f32_16x16x4_f32(false, a2, false, b0, (short)0, acc20, false, false);
            acc30 = __builtin_amdgcn_wmma_f32_16x16x4_f32(false, a3, false, b0, (short)0, acc30, false, false);
            acc01 = __builtin_amdgcn_wmma_f32_16x16x4_f32(false, a0, false, b1, (short)0, acc01, false, false);
            acc11 = __builtin_amdgcn_wmma_f32_16x16x4_f32(false, a1, false, b1, (short)0, acc11, false, false);
            acc21 = __builtin_amdgcn_wmma_f32_16x16x4_f32(false, a2, false, b1, (short)0, acc21, false, false);
            acc31 = __builtin_amdgcn_wmma_f32_16x16x4_f32(false, a3, false, b1, (short)0, acc31, false, false);
        }
    }

    // Epilogue: C/D layout -> VGPR r holds M=r (lanes 0-15) / M=r+8 (lanes 16-31).
    const int rowoff = (lane >> 4) * 8;
    const bool st1 = (ncol1 < NCOLS);
    v8f t0[4] = {acc00, acc10, acc20, acc30};
    v8f t1[4] = {acc01, acc11, acc21, acc31};
#pragma unroll
    for (int s = 0; s < 4; ++s) {
        const int vbase = mblockbase + wave * MWAVE + s * 16 + rowoff;
#pragma unroll
        for (int r = 0; r < 8; ++r) {
            const long vo = (long)(vbase + r) * OUTVOX;
            out[vo + colofs0] = t0[s][r];
            if (st1) out[vo + colofs1] = t1[s][r];
        }
    }
}

// ---------------------------------------------------------------------------
// Mean column + fixed entries. One block (64 threads) per voxel.
//   out[v][0][0]   = chat0[v]
//   out[v][0][1]   = 0.01          (cov[0,0])
//   out[v][0][2+b] = 0             (cov[0,1+b])
//   out[v][1+a][0] = (X @ Wm)[v,a]
//   out[v][1+a][1] = 0             (cov[1+a,0])
// ---------------------------------------------------------------------------
__global__ __launch_bounds__(64)
void mean_fill(const float* __restrict__ X,
               const float* __restrict__ wm,     // vec_W_post_mean, (R*KM1)
               const float* __restrict__ chat0,
               float* __restrict__ out)
{
    const int v = blockIdx.x;
    const int t = threadIdx.x;
    __shared__ float xr[RR];
    xr[t] = X[v * RR + t];
    __syncthreads();

    float* o = out + (long)v * OUTVOX;
    if (t < KM1) {
        float s = 0.f;
#pragma unroll
        for (int i = 0; i < RR; ++i) s += xr[i] * wm[i * KM1 + t];
        o[(t + 1) * OUTROW + 0] = s;      // mean column
        o[2 + t]                = 0.f;    // cov row 0
        o[(t + 1) * OUTROW + 1] = 0.f;    // cov col 0
    } else if (t == KM1) {
        o[0] = chat0[v];
    } else if (t == KM1 + 1) {
        o[1] = 0.01f;                     // SIGMA2_MU
    }
}

extern "C" void kernel_launch(void* const* d_in, const int* in_sizes, int n_in,
                              void* d_out, int out_size, void* d_ws, size_t ws_size,
                              hipStream_t stream)
{
    (void)in_sizes; (void)n_in; (void)out_size; (void)d_ws; (void)ws_size;
    const float* X     = (const float*)d_in[0];   // (8192, 64)
    const float* Wm    = (const float*)d_in[1];   // (2816,)
    const float* C     = (const float*)d_in[2];   // (2816, 2816)
    const float* chat0 = (const float*)d_in[3];   // (8192, 1)
    float* out = (float*)d_out;                   // (8192, 45, 46)

    dim3 grid(NGRID /* 61 */, NVOX / MBLOCK /* 64 */);
    cov_gemm<<<grid, BLOCK, 0, stream>>>(X, C, out);
    mean_fill<<<NVOX, 64, 0, stream>>>(X, Wm, chat0, out);
}